// STDSLReLU_72842645340570
// MI455X (gfx1250) — compile-verified
//
#include <hip/hip_runtime.h>

// Fused 10-iteration depthwise-Gaussian + smoothed-LeakyReLU solver.
// One read of o, one write of u. 84x88 fp32 staging region per 64x64 tile,
// staged with CDNA5 async global->LDS copies (ASYNCcnt) into 320KB WGP LDS.
// Middle iterations use column-rolling registers: 3 LDS loads + 1 store per
// cell (vs 9+1 naive), unit-stride across lanes -> bank-conflict-free.

#define TILE   64
#define HALO   10
#define RROWS  84            // TILE + 2*HALO
#define RS     88            // region width / LDS row stride (16B aligned)
#define XOFF   12            // region col 0 -> global x = tx*64 - 12 (4-aligned for B128)
#define YOFF   10            // region row 0 -> global y = ty*64 - 10
#define IMG_W  256
#define IMG_H  256

__device__ __forceinline__ float slrelu(float x) {
    // 0.2*x + 0.8*softplus(10x)/10, stable softplus via max + log1p(exp(-|y|))
    float y  = 10.0f * x;
    float sp = fmaxf(y, 0.0f) + __logf(1.0f + __expf(-fabsf(y)));
    return 0.2f * x + 0.08f * sp;
}

__global__ __launch_bounds__(256)
void stdslrelu_fused_kernel(const float* __restrict__ o,
                            const float* __restrict__ nb_sigma,
                            const float* __restrict__ lam_p,
                            float* __restrict__ out) {
    extern __shared__ float smem[];
    float* o_s = smem;                    // [RROWS*RS] staged o tile (+halo)
    float* vA  = smem + RROWS * RS;       // ping: v = 1 - 2u
    float* vB  = smem + 2 * RROWS * RS;   // pong

    const int tid = threadIdx.x;
    const int wg  = blockIdx.x;
    const int tx  = wg & 3;               // 4 tiles in x
    const int ty  = (wg >> 2) & 3;        // 4 tiles in y
    const int img = wg >> 4;              // n*64 + channel, 512 images
    const int ch  = img & 63;

    const int x0 = tx * TILE - XOFF;      // global col of region col 0
    const int y0 = ty * TILE - YOFF;      // global row of region row 0

    // ---- stage o tile (with halo) via async global->LDS B128 copies ----
    const float* obase = o + (size_t)img * (IMG_W * IMG_H);
    for (int j = tid; j < RROWS * (RS / 4); j += 256) {
        int row = j / (RS / 4);
        int ck  = j % (RS / 4);
        int gy  = min(max(y0 + row, 0), IMG_H - 1);     // clamp: halo cells masked later
        int gx  = min(max(x0 + 4 * ck, 0), IMG_W - 4);  // chunks never straddle the border
        const float* gp = obase + gy * IMG_W + gx;
        unsigned lds = (unsigned)(unsigned long long)(const void*)&o_s[row * RS + 4 * ck];
        asm volatile("global_load_async_to_lds_b128 %0, %1, off"
                     :: "v"(lds), "v"(gp) : "memory");
    }

    // ---- overlap: uniform per-channel Gaussian weights + lambda ----
    float sig  = fmaxf(nb_sigma[ch], 1e-6f);
    float is2  = 1.0f / (2.0f * sig * sig);
    float w1   = __expf(-is2);            // r^2 = 1
    float w2   = w1 * w1;                 // r^2 = 2
    float invn = 1.0f / (1.0f + 4.0f * w1 + 4.0f * w2 + 1e-12f);
    float wc = invn, we = w1 * invn, wd = w2 * invn;
    float lam = lam_p[0];

    asm volatile("s_wait_asynccnt 0x0" ::: "memory");
    __syncthreads();

    // ---- v0 = 1 - 2*slrelu(o), zero outside the image (conv zero-padding) ----
    for (int idx = tid; idx < RROWS * RS; idx += 256) {
        int r = idx / RS, c = idx % RS;
        bool in = ((unsigned)(y0 + r) < (unsigned)IMG_H) &
                  ((unsigned)(x0 + c) < (unsigned)IMG_W);
        float u0 = slrelu(o_s[idx]);
        vA[idx] = in ? (1.0f - 2.0f * u0) : 0.0f;
    }
    __syncthreads();

    // ---- iterations 1..9: column-rolling 9-point stencil on shrinking regions ----
    float* vsrc = vA;
    float* vdst = vB;
#pragma unroll
    for (int t = 1; t <= 9; ++t) {
        const int w    = RROWS - 2 * t;        // active width == active height (66..82)
        const int nseg = 256 / w;              // row-segments per column (==3 here)
        if (tid < nseg * w) {
            const int c   = 2 + t + (tid % w); // this thread's column
            const int seg = tid / w;
            const int r0  = t + (seg * w) / nseg;
            const int r1  = t + ((seg + 1) * w) / nseg;
            const bool colIn = ((unsigned)(x0 + c) < (unsigned)IMG_W);

            int b = (r0 - 1) * RS + c;                         // row r0-1
            float a0 = vsrc[b - 1], a1 = vsrc[b], a2 = vsrc[b + 1];
            b += RS;                                           // row r0
            float b0 = vsrc[b - 1], b1 = vsrc[b], b2 = vsrc[b + 1];

            for (int r = r0; r < r1; ++r) {
                const int bn = b + RS;                         // row r+1
                float c0 = vsrc[bn - 1], c1 = vsrc[bn], c2 = vsrc[bn + 1];
                float q = wc * b1
                        + we * (a1 + c1 + b0 + b2)
                        + wd * (a0 + a2 + c0 + c2);
                float u = slrelu(o_s[b] - lam * q);
                bool in = colIn & ((unsigned)(y0 + r) < (unsigned)IMG_H);
                vdst[b] = in ? (1.0f - 2.0f * u) : 0.0f;
                a0 = b0; a1 = b1; a2 = b2;
                b0 = c0; b1 = c1; b2 = c2;
                b = bn;
            }
        }
        __syncthreads();
        float* tp = vsrc; vsrc = vdst; vdst = tp;
    }

    // ---- iteration 10: column-rolling, write u directly to global (all in-image) ----
    {
        const int c   = XOFF + (tid & 63);                 // 64 columns x 4 segments
        const int seg = tid >> 6;
        const int r0  = YOFF + seg * 16;

        float* outp = out + (size_t)img * (IMG_W * IMG_H)
                    + (ty * TILE + seg * 16) * IMG_W + (tx * TILE + (tid & 63));

        int b = (r0 - 1) * RS + c;
        float a0 = vsrc[b - 1], a1 = vsrc[b], a2 = vsrc[b + 1];
        b += RS;
        float b0 = vsrc[b - 1], b1 = vsrc[b], b2 = vsrc[b + 1];

        for (int i = 0; i < 16; ++i) {
            const int bn = b + RS;
            float c0 = vsrc[bn - 1], c1 = vsrc[bn], c2 = vsrc[bn + 1];
            float q = wc * b1
                    + we * (a1 + c1 + b0 + b2)
                    + wd * (a0 + a2 + c0 + c2);
            outp[i * IMG_W] = slrelu(o_s[b] - lam * q);
            a0 = b0; a1 = b1; a2 = b2;
            b0 = c0; b1 = c1; b2 = c2;
            b = bn;
        }
    }
}

extern "C" void kernel_launch(void* const* d_in, const int* in_sizes, int n_in,
                              void* d_out, int out_size, void* d_ws, size_t ws_size,
                              hipStream_t stream) {
    const float* o   = (const float*)d_in[0];   // [8,64,256,256] f32
    const float* sig = (const float*)d_in[1];   // [64,1,1] f32
    const float* lam = (const float*)d_in[2];   // [1] f32
    float* out = (float*)d_out;                 // [8,64,256,256] f32

    const size_t lds_bytes = 3u * RROWS * RS * sizeof(float); // 88,704 B / workgroup
    // 8 n * 64 c * (4x4) tiles = 8192 workgroups, 256 threads (8 wave32s) each
    stdslrelu_fused_kernel<<<8192, 256, lds_bytes, stream>>>(o, sig, lam, out);
}